// GAT_50861002719407
// MI455X (gfx1250) — compile-verified
//
#include <hip/hip_runtime.h>
#include <hip/hip_bf16.h>

#define GAT_N 30000
#define GAT_E 480000
#define GAT_ETOT (GAT_E + GAT_N)   // edges + self loops
#define GAT_H 8
#define GAT_C 32
#define GAT_FH 256
#define NEG_SLOPE 0.2f

typedef float v2f __attribute__((ext_vector_type(2)));
typedef float v8f __attribute__((ext_vector_type(8)));

// ---------------------------------------------------------------------------
// GEMM: C[M x Ncols] = A[M x K] @ B[K x Ncols] using native fp32 WMMA
// V_WMMA_F32_16X16X4_F32. One wave computes one 16x16 tile; 4 waves/block
// cover 4 adjacent N-tiles (same A rows -> L2 reuse). M, Ncols multiples of
// 16; K multiple of 4. EXEC is all-ones (exact grid fit) as WMMA requires.
//
// f32 A 16x4 fragment (ISA 7.12.2): lanes 0-15 -> M=lane, v[0]=K0,v[1]=K1;
//                                   lanes 16-31 -> M=lane-16, v[0]=K2,v[1]=K3.
// f32 B 4x16 fragment (mirror):     lanes 0-15 -> N=lane, v[0]=K0,v[1]=K1;
//                                   lanes 16-31 -> N=lane-16, v[0]=K2,v[1]=K3.
// f32 C/D 16x16: VGPR r: lanes 0-15 -> M=r, N=lane; lanes 16-31 -> M=8+r.
// ---------------------------------------------------------------------------
__global__ void gemm_wmma_f32(const float* __restrict__ A,
                              const float* __restrict__ B,
                              float* __restrict__ C,
                              int M, int K, int Ncols) {
  const int wave = threadIdx.x >> 5;
  const int lane = threadIdx.x & 31;
  const int half = lane >> 4;     // 0: K0/K1 side, 1: K2/K3 side
  const int l    = lane & 15;

  const int m0 = blockIdx.x * 16;
  const int n0 = blockIdx.y * 64 + wave * 16;
  if (m0 >= M || n0 >= Ncols) return;

  v8f c = {0.f, 0.f, 0.f, 0.f, 0.f, 0.f, 0.f, 0.f};

  const float* arow = A + (size_t)(m0 + l) * K + half * 2;  // A[m][k + 2*half]
  const float* bcol = B + (size_t)(half * 2) * Ncols + n0 + l;

  for (int k = 0; k < K; k += 4) {
    v2f a = *(const v2f*)(arow + k);      // {A[m][k+2h], A[m][k+2h+1]}
    v2f b;
    b.x = bcol[(size_t)k * Ncols];        // B[k+2h  ][n]
    b.y = bcol[(size_t)k * Ncols + Ncols];// B[k+2h+1][n]
    c = __builtin_amdgcn_wmma_f32_16x16x4_f32(
        /*neg_a=*/false, a, /*neg_b=*/false, b,
        /*c_mod=*/(short)0, c, /*reuse_a=*/false, /*reuse_b=*/false);
  }

  float* crow = C + n0 + l;
  const int mbase = m0 + half * 8;
#pragma unroll
  for (int r = 0; r < 8; ++r)
    crow[(size_t)(mbase + r) * Ncols] = c[r];
}

// alpha_s[n,h] = sum_c h[n,h,c]*a_src[h,c]; alpha_d likewise. 1 thread/(n,h).
__global__ void alphas_kernel(const float* __restrict__ h,
                              const float* __restrict__ asrc,
                              const float* __restrict__ adst,
                              float* __restrict__ s, float* __restrict__ d,
                              int total) {
  int i = blockIdx.x * blockDim.x + threadIdx.x;
  if (i >= total) return;
  int node = i >> 3, hd = i & 7;
  const float* hp = h + (size_t)node * GAT_FH + hd * GAT_C;
  const float* ap = asrc + hd * GAT_C;
  const float* bp = adst + hd * GAT_C;
  float ss = 0.f, dd = 0.f;
#pragma unroll
  for (int cc = 0; cc < GAT_C; ++cc) {
    float v = hp[cc];
    ss += v * ap[cc];
    dd += v * bp[cc];
  }
  s[i] = ss;
  d[i] = dd;
}

// zero agg[N*FH], set emax[N*H] = -inf-ish, denom[N*H] = 0
__global__ void init_kernel(float* __restrict__ agg, float* __restrict__ emax,
                            float* __restrict__ denom, int nfeat, int nh) {
  int i = blockIdx.x * blockDim.x + threadIdx.x;
  if (i < nfeat) agg[i] = 0.f;
  if (i < nh) { emax[i] = -3.0e38f; denom[i] = 0.f; }
}

__device__ __forceinline__ void edge_endpoints(const int* __restrict__ ei,
                                               int e, int& src, int& dst) {
  if (e < GAT_E) { src = ei[e]; dst = ei[GAT_E + e]; }
  else           { src = e - GAT_E; dst = e - GAT_E; }   // self loop
}

// segment max of leaky_relu(alpha_s[src]+alpha_d[dst]) into emax[dst,h]
__global__ void edge_max_kernel(const int* __restrict__ ei,
                                const float* __restrict__ as,
                                const float* __restrict__ ad,
                                float* __restrict__ emax) {
  int e = blockIdx.x * blockDim.x + threadIdx.x;
  if (e >= GAT_ETOT) return;
  int src, dst;
  edge_endpoints(ei, e, src, dst);
#pragma unroll
  for (int hd = 0; hd < GAT_H; ++hd) {
    float v = as[src * GAT_H + hd] + ad[dst * GAT_H + hd];
    v = v > 0.f ? v : NEG_SLOPE * v;
    atomicMax(&emax[dst * GAT_H + hd], v);   // GLOBAL_ATOMIC_MAX_NUM_F32
  }
}

// denom[dst,h] += exp(e - emax[dst,h])
__global__ void edge_denom_kernel(const int* __restrict__ ei,
                                  const float* __restrict__ as,
                                  const float* __restrict__ ad,
                                  const float* __restrict__ emax,
                                  float* __restrict__ denom) {
  int e = blockIdx.x * blockDim.x + threadIdx.x;
  if (e >= GAT_ETOT) return;
  int src, dst;
  edge_endpoints(ei, e, src, dst);
#pragma unroll
  for (int hd = 0; hd < GAT_H; ++hd) {
    float v = as[src * GAT_H + hd] + ad[dst * GAT_H + hd];
    v = v > 0.f ? v : NEG_SLOPE * v;
    atomicAdd(&denom[dst * GAT_H + hd], __expf(v - emax[dst * GAT_H + hd]));
  }
}

// agg[dst,h,:] += alpha * h[src,h,:]   (1 thread per (edge,head), float4 gather)
__global__ void edge_agg_kernel(const int* __restrict__ ei,
                                const float* __restrict__ as,
                                const float* __restrict__ ad,
                                const float* __restrict__ emax,
                                const float* __restrict__ denom,
                                const float* __restrict__ h,
                                float* __restrict__ agg) {
  int t = blockIdx.x * blockDim.x + threadIdx.x;
  if (t >= GAT_ETOT * GAT_H) return;
  int e = t >> 3, hd = t & 7;
  int src, dst;
  edge_endpoints(ei, e, src, dst);
  float v = as[src * GAT_H + hd] + ad[dst * GAT_H + hd];
  v = v > 0.f ? v : NEG_SLOPE * v;
  int di = dst * GAT_H + hd;
  float alpha = __expf(v - emax[di]) / (denom[di] + 1e-16f);
  const float4* hp = (const float4*)(h + (size_t)src * GAT_FH + hd * GAT_C);
  float* op = agg + (size_t)dst * GAT_FH + hd * GAT_C;
#pragma unroll
  for (int q = 0; q < GAT_C / 4; ++q) {
    float4 x4 = hp[q];
    atomicAdd(op + q * 4 + 0, x4.x * alpha);
    atomicAdd(op + q * 4 + 1, x4.y * alpha);
    atomicAdd(op + q * 4 + 2, x4.z * alpha);
    atomicAdd(op + q * 4 + 3, x4.w * alpha);
  }
}

// in-place: agg = elu(agg + bias)
__global__ void bias_elu_kernel(float* __restrict__ agg,
                                const float* __restrict__ b, int total) {
  int i = blockIdx.x * blockDim.x + threadIdx.x;
  if (i >= total) return;
  float v = agg[i] + b[i & (GAT_FH - 1)];
  agg[i] = v > 0.f ? v : (__expf(v) - 1.f);
}

// y[n] = h[n,:] . Wl + bl  — one wave32 per node, shuffle reduction
__global__ void final_linear_kernel(const float* __restrict__ h,
                                    const float* __restrict__ Wl,
                                    const float* __restrict__ bl,
                                    float* __restrict__ y, int n) {
  int gid = blockIdx.x * blockDim.x + threadIdx.x;
  int wid = gid >> 5;
  int lane = threadIdx.x & 31;
  if (wid >= n) return;
  const float* hp = h + (size_t)wid * GAT_FH;
  float s = 0.f;
#pragma unroll
  for (int c = lane; c < GAT_FH; c += 32) s += hp[c] * Wl[c];
#pragma unroll
  for (int off = 16; off > 0; off >>= 1) s += __shfl_down(s, off, 32);
  if (lane == 0) y[wid] = s + bl[0];
}

// ---------------------------------------------------------------------------
extern "C" void kernel_launch(void* const* d_in, const int* in_sizes, int n_in,
                              void* d_out, int out_size, void* d_ws, size_t ws_size,
                              hipStream_t stream) {
  const float* x      = (const float*)d_in[0];
  const int*   ei     = (const int*)  d_in[1];
  const float* W1     = (const float*)d_in[2];
  const float* a1_src = (const float*)d_in[3];
  const float* a1_dst = (const float*)d_in[4];
  const float* b1     = (const float*)d_in[5];
  const float* W2     = (const float*)d_in[6];
  const float* a2_src = (const float*)d_in[7];
  const float* a2_dst = (const float*)d_in[8];
  const float* b2     = (const float*)d_in[9];
  const float* Wl     = (const float*)d_in[10];
  const float* bl     = (const float*)d_in[11];
  float* out = (float*)d_out;

  // workspace layout (floats): hbuf[N*FH] | agg[N*FH] | emax[N*H] | denom[N*H] | as[N*H] | ad[N*H]
  float* hbuf  = (float*)d_ws;
  float* agg   = hbuf  + (size_t)GAT_N * GAT_FH;
  float* emax  = agg   + (size_t)GAT_N * GAT_FH;
  float* denom = emax  + (size_t)GAT_N * GAT_H;
  float* as    = denom + (size_t)GAT_N * GAT_H;
  float* ad    = as    + (size_t)GAT_N * GAT_H;

  const int NF  = GAT_N * GAT_FH;   // 7,680,000
  const int NH  = GAT_N * GAT_H;    // 240,000
  const int EH  = GAT_ETOT * GAT_H; // 4,080,000

  dim3 gemm_grid(GAT_N / 16, GAT_FH / 64);   // 1875 x 4, exact fit
  dim3 gemm_blk(128);                        // 4 waves (wave32)

  const int TPB = 256;
  auto blocks = [](int n, int tpb) { return (n + tpb - 1) / tpb; };

  // ---------------- Layer 1 ----------------
  gemm_wmma_f32<<<gemm_grid, gemm_blk, 0, stream>>>(x, W1, hbuf, GAT_N, 128, GAT_FH);
  alphas_kernel<<<blocks(NH, TPB), TPB, 0, stream>>>(hbuf, a1_src, a1_dst, as, ad, NH);
  init_kernel<<<blocks(NF, TPB), TPB, 0, stream>>>(agg, emax, denom, NF, NH);
  edge_max_kernel<<<blocks(GAT_ETOT, TPB), TPB, 0, stream>>>(ei, as, ad, emax);
  edge_denom_kernel<<<blocks(GAT_ETOT, TPB), TPB, 0, stream>>>(ei, as, ad, emax, denom);
  edge_agg_kernel<<<blocks(EH, TPB), TPB, 0, stream>>>(ei, as, ad, emax, denom, hbuf, agg);
  bias_elu_kernel<<<blocks(NF, TPB), TPB, 0, stream>>>(agg, b1, NF);
  // agg now holds layer-1 activation (input to layer 2)

  // ---------------- Layer 2 ----------------
  gemm_wmma_f32<<<gemm_grid, gemm_blk, 0, stream>>>(agg, W2, hbuf, GAT_N, GAT_FH, GAT_FH);
  alphas_kernel<<<blocks(NH, TPB), TPB, 0, stream>>>(hbuf, a2_src, a2_dst, as, ad, NH);
  init_kernel<<<blocks(NF, TPB), TPB, 0, stream>>>(agg, emax, denom, NF, NH);
  edge_max_kernel<<<blocks(GAT_ETOT, TPB), TPB, 0, stream>>>(ei, as, ad, emax);
  edge_denom_kernel<<<blocks(GAT_ETOT, TPB), TPB, 0, stream>>>(ei, as, ad, emax, denom);
  edge_agg_kernel<<<blocks(EH, TPB), TPB, 0, stream>>>(ei, as, ad, emax, denom, hbuf, agg);
  bias_elu_kernel<<<blocks(NF, TPB), TPB, 0, stream>>>(agg, b2, NF);

  // ---------------- Final linear ----------------
  final_linear_kernel<<<blocks(GAT_N * 32, TPB), TPB, 0, stream>>>(agg, Wl, bl, out, GAT_N);
}